// QFCModel_65481071410040
// MI455X (gfx1250) — compile-verified
//
#include <hip/hip_runtime.h>
#include <math.h>

typedef float v2f __attribute__((ext_vector_type(2)));
typedef float v8f __attribute__((ext_vector_type(8)));

#define SPB 128          // samples per block (8 waves x 16 samples)
#define THREADS_A 256

// ---------------------------------------------------------------------------
// Kernel A: stream rows 0-5 of each image, 6x6 avg-pool via f32 WMMA,
// closed-form 4-qubit circuit, per-block deterministic stats partials.
// ---------------------------------------------------------------------------
__global__ __launch_bounds__(THREADS_A)
void pool_circuit_kernel(const float* __restrict__ x,
                         const float* __restrict__ params,
                         float* __restrict__ zbuf,
                         float* __restrict__ partials,
                         int B)
{
    __shared__ float pooled[SPB][4];   // pooled means per sample
    __shared__ float red[SPB][8];      // {z0..z3, z0^2..z3^2} tree reduction
    __shared__ float sAB[8];           // A_j = cos(beta_j); B_j = sin(beta_j)*sin(alpha_j)

    const int tid   = threadIdx.x;
    const int lane  = tid & 31;
    const int wv    = tid >> 5;        // wave id 0..7
    const int nIdx  = lane & 15;       // M (A) / N (B) index within half
    const int kBase = (lane >> 4) << 1; // lanes 0-15 -> K{0,1}; lanes 16-31 -> K{2,3}

    if (tid < 4) {
        float al = params[2 * tid + 0];
        float be = params[2 * tid + 1];
        sAB[tid]     = cosf(be);
        sAB[tid + 4] = sinf(be) * sinf(al);
    }

    // B-operand patterns for the 6 column-chunks of one image row.
    // Column c (0..23) belongs to pooled block c/6; weight 1/36 folds the mean.
    v2f bt[6];
#pragma unroll
    for (int t = 0; t < 6; ++t) {
        int c0 = 4 * t + kBase;
        bt[t].x = (nIdx == (c0    ) / 6) ? (1.0f / 36.0f) : 0.0f;
        bt[t].y = (nIdx == (c0 + 1) / 6) ? (1.0f / 36.0f) : 0.0f;
    }

    const int sampleBase = blockIdx.x * SPB + wv * 16;
    // A-operand source: sample (base + nIdx), this lane's K-half
    const float* px = x + (size_t)(sampleBase + nIdx) * 576 + kBase;

    v8f acc = {0.f, 0.f, 0.f, 0.f, 0.f, 0.f, 0.f, 0.f};
#pragma unroll
    for (int row = 0; row < 6; ++row) {
#pragma unroll
        for (int t = 0; t < 6; ++t) {
            v2f a = *(const v2f*)(px + row * 24 + t * 4);
            // D += A(16x4) * B(4x16); accumulates the 6x6 block sums in f32
            acc = __builtin_amdgcn_wmma_f32_16x16x4_f32(
                false, a, false, bt[t], (short)0, acc, false, false);
        }
    }

    // D layout: VGPR r -> sample M=r (lanes 0-15) or r+8 (lanes 16-31); N = nIdx
    if (nIdx < 4) {
        int srow = wv * 16 + ((lane >> 4) << 3);
#pragma unroll
        for (int r = 0; r < 8; ++r)
            pooled[srow + r][nIdx] = acc[r];
    }
    __syncthreads();

    // Closed-form circuit: q_j = cos(b_j)cos(x_j) + sin(b_j)sin(a_j)sin(x_j),
    // <Z_i> = prod_{j<=i} q_j  (CNOT chain is a basis permutation of a product state)
    if (tid < SPB) {
        float z[4];
        float prod = 1.0f;
#pragma unroll
        for (int j = 0; j < 4; ++j) {
            float xj = pooled[tid][j];
            float sx, cx;
            sincosf(xj, &sx, &cx);
            prod *= sAB[j] * cx + sAB[4 + j] * sx;
            z[j] = prod;
        }
        size_t s = (size_t)blockIdx.x * SPB + tid;
        *(float4*)(zbuf + s * 4) = make_float4(z[0], z[1], z[2], z[3]);
#pragma unroll
        for (int j = 0; j < 4; ++j) {
            red[tid][j]     = z[j];
            red[tid][4 + j] = z[j] * z[j];
        }
    }
    __syncthreads();

    // fixed-order binary tree -> deterministic batch stats
    for (int s = SPB / 2; s > 0; s >>= 1) {
        if (tid < s) {
#pragma unroll
            for (int j = 0; j < 8; ++j) red[tid][j] += red[tid + s][j];
        }
        __syncthreads();
    }
    if (tid < 8) partials[(size_t)blockIdx.x * 8 + tid] = red[0][tid];
}

// ---------------------------------------------------------------------------
// Kernel B: fold block partials -> per-channel scale/shift (training BN)
// ---------------------------------------------------------------------------
__global__ void stats_kernel(const float* __restrict__ partials, int nblk, int B,
                             const float* __restrict__ gamma,
                             const float* __restrict__ beta,
                             float* __restrict__ stats)
{
    __shared__ float sums[8];
    int j = threadIdx.x;
    if (j < 8) {
        float s = 0.f;
        for (int b = 0; b < nblk; ++b) s += partials[(size_t)b * 8 + j];
        sums[j] = s;
    }
    __syncthreads();
    if (j < 4) {
        float invB = 1.0f / (float)B;
        float mean = sums[j] * invB;
        float var  = sums[j + 4] * invB - mean * mean;   // biased variance
        float sc   = gamma[j] * rsqrtf(var + 1e-5f);
        stats[j]     = sc;
        stats[4 + j] = beta[j] - mean * sc;
    }
}

// ---------------------------------------------------------------------------
// Kernel C: out = z * scale + shift   (vectorized, 2 MB of traffic)
// ---------------------------------------------------------------------------
__global__ __launch_bounds__(256)
void norm_kernel(const float* __restrict__ zbuf,
                 const float* __restrict__ stats,
                 float* __restrict__ out, int B)
{
    int i = blockIdx.x * blockDim.x + threadIdx.x;
    if (i < B) {
        float4 z  = *(const float4*)(zbuf + (size_t)i * 4);
        float4 sc = *(const float4*)(stats);
        float4 sh = *(const float4*)(stats + 4);
        float4 o;
        o.x = z.x * sc.x + sh.x;
        o.y = z.y * sc.y + sh.y;
        o.z = z.z * sc.z + sh.z;
        o.w = z.w * sc.w + sh.w;
        *(float4*)(out + (size_t)i * 4) = o;
    }
}

extern "C" void kernel_launch(void* const* d_in, const int* in_sizes, int n_in,
                              void* d_out, int out_size, void* d_ws, size_t ws_size,
                              hipStream_t stream)
{
    const float* x      = (const float*)d_in[0];   // [B,1,24,24]
    const float* params = (const float*)d_in[1];   // [4,2]
    const float* gamma  = (const float*)d_in[2];   // [4]
    const float* beta   = (const float*)d_in[3];   // [4]
    float* out = (float*)d_out;                    // [B,4]

    int B = in_sizes[0] / 576;                     // 65536
    int nblkA = B / SPB;                           // 512 (B is a multiple of 128)

    // workspace layout: zbuf[B*4] | partials[nblkA*8] | stats[8]
    float* zbuf     = (float*)d_ws;
    float* partials = zbuf + (size_t)B * 4;
    float* stats    = partials + (size_t)nblkA * 8;

    pool_circuit_kernel<<<nblkA, THREADS_A, 0, stream>>>(x, params, zbuf, partials, B);
    stats_kernel<<<1, 32, 0, stream>>>(partials, nblkA, B, gamma, beta, stats);
    norm_kernel<<<(B + 255) / 256, 256, 0, stream>>>(zbuf, stats, out, B);
}